// Model_70403103916230
// MI455X (gfx1250) — compile-verified
//
#include <hip/hip_runtime.h>
#include <hip/hip_bf16.h>

// Model: grid-embedding -> LSTM(32 steps) -> BatchNorm -> proj -> batched proj . tb^T
// B=8, L=32, N=256, M=256.  All fp32, all GEMMs on v_wmma_f32_16x16x4_f32.
// Refactor: z_x = X@(E1@W1^T) + (X.w)@(E2@W2^T)  (G1,G2 precomputed, 256x1024),
// so the per-step gate GEMM is a single K=768 accumulation: X@G1 + Xw@G2 + h@Whh^T.

#define DI __device__ __forceinline__

typedef float v2f __attribute__((ext_vector_type(2)));
typedef float v8f __attribute__((ext_vector_type(8)));

constexpr int LDA = 36; // 64x32 A tile + pad; r*36 mod 64 = 4*(9r mod 16): conflict-free frag reads
constexpr int LDB = 65; // 32x64 B tile + pad; stride-65 transpose stores are conflict-free

// ---- wave-level 16x16x4 f32 WMMA tile engine (layouts per CDNA5 ISA 7.12.2) ----
DI v2f frag_a(const float* As, int k0, int r, int half) {
  v2f a;
  a.x = As[r * LDA + k0 + 2 * half + 0];
  a.y = As[r * LDA + k0 + 2 * half + 1];
  return a;
}
DI v2f frag_b(const float* Bs, int k0, int r, int half) {
  v2f b;
  b.x = Bs[(k0 + 2 * half + 0) * LDB + r];
  b.y = Bs[(k0 + 2 * half + 1) * LDB + r];
  return b;
}

// Process one 32-deep K chunk: wave 'wr' owns rows [16*wr,16*wr+16), col-pair 'cp'
// owns cols [32*cp,32*cp+32) as two 16-wide accumulators sharing the A fragment.
DI void mma_chunk(const float* As, const float* Bs, int wr, int cp, int r, int half,
                  v8f& acc0, v8f& acc1) {
  const float* Aw = As + (wr * 16) * LDA;
  const float* B0 = Bs + 32 * cp;
  const float* B1 = Bs + 32 * cp + 16;
#pragma unroll
  for (int k0 = 0; k0 < 32; k0 += 4) {
    v2f a  = frag_a(Aw, k0, r, half);
    v2f b0 = frag_b(B0, k0, r, half);
    v2f b1 = frag_b(B1, k0, r, half);
    acc0 = __builtin_amdgcn_wmma_f32_16x16x4_f32(false, a, false, b0, (short)0, acc0, false, false);
    acc1 = __builtin_amdgcn_wmma_f32_16x16x4_f32(false, a, false, b1, (short)0, acc1, false, false);
  }
}

// D layout: VGPR i -> row i (lanes 0-15) / row i+8 (lanes 16-31), col = lane&15.
DI void store_tile(float* out, int ldo, const v8f& acc, int r, int half) {
#pragma unroll
  for (int i = 0; i < 8; ++i) out[(i + 8 * half) * ldo + r] = acc[i];
}

#define WMMA_THREAD_DECOMP()        \
  const int tid  = threadIdx.x;     \
  const int lane = tid & 31;        \
  const int wave = tid >> 5;        \
  const int half = lane >> 4;       \
  const int r    = lane & 15;       \
  const int wr   = wave & 3;        \
  const int cp   = wave >> 2;       \
  (void)lane;

DI float sigmoidf(float x) { return 1.0f / (1.0f + __expf(-x)); }

// ---- K0: w = softmax(-dis, axis=-1), one block per row ----
__global__ void softmax_rows(const float* __restrict__ dis, float* __restrict__ w) {
  __shared__ float red[256];
  const int i = blockIdx.x, t = threadIdx.x;
  float v = -dis[i * 256 + t];
  red[t] = v;
  __syncthreads();
  for (int s = 128; s > 0; s >>= 1) {
    if (t < s) red[t] = fmaxf(red[t], red[t + s]);
    __syncthreads();
  }
  float mx = red[0];
  __syncthreads();
  float e = __expf(v - mx);
  red[t] = e;
  __syncthreads();
  for (int s = 128; s > 0; s >>= 1) {
    if (t < s) red[t] += red[t + s];
    __syncthreads();
  }
  w[i * 256 + t] = e / red[0];
}

// ---- K1: G{1,2}[d][c] = sum_m E{1,2}[d][m] * W_ih[c][m (+256)]  (256 x 1024, K=256) ----
__global__ void embed_weight_gemm(const float* __restrict__ E1, const float* __restrict__ E2,
                                  const float* __restrict__ Wih,
                                  float* __restrict__ G1, float* __restrict__ G2) {
  __shared__ float As[64 * LDA];
  __shared__ float Bs[32 * LDB];
  WMMA_THREAD_DECOMP();
  const float* E = blockIdx.z ? E2 : E1;
  float* G = blockIdx.z ? G2 : G1;
  const int koff = blockIdx.z ? 256 : 0;
  const int rb = blockIdx.y * 64;
  const int cb = blockIdx.x * 64;
  v8f acc0 = {}; v8f acc1 = {};
  for (int k0 = 0; k0 < 256; k0 += 32) {
    __syncthreads();
#pragma unroll
    for (int i = 0; i < 8; ++i) {
      int idx = tid + i * 256;
      int ar = idx >> 5, ac = idx & 31;
      As[ar * LDA + ac] = E[(rb + ar) * 256 + k0 + ac];
      int bc = idx >> 5, bk = idx & 31;                      // B[k][c] = W[c][k] (transpose load)
      Bs[bk * LDB + bc] = Wih[(cb + bc) * 512 + koff + k0 + bk];
    }
    __syncthreads();
    mma_chunk(As, Bs, wr, cp, r, half, acc0, acc1);
  }
  float* o0 = G + (rb + 16 * wr) * 1024 + cb + 32 * cp;
  store_tile(o0, 1024, acc0, r, half);
  store_tile(o0 + 16, 1024, acc1, r, half);
}

// ---- K2 (per step l): z[(b,n)][c] = X_l@G1 + (X_l.w)@G2 + h@Whh^T   (2048 x 1024, K=768) ----
__global__ void lstm_gemm(const float* __restrict__ X, const float* __restrict__ wsm,
                          const float* __restrict__ G1, const float* __restrict__ G2,
                          const float* __restrict__ Whh, const float* __restrict__ h,
                          float* __restrict__ z, int l) {
  __shared__ float As[64 * LDA];
  __shared__ float Bs[32 * LDB];
  WMMA_THREAD_DECOMP();
  const int rb = blockIdx.y * 64;
  const int cb = blockIdx.x * 64;
  v8f acc0 = {}; v8f acc1 = {};
  for (int seg = 0; seg < 2; ++seg) {          // X@G1 then (X.w)@G2
    const float* G = seg ? G2 : G1;
    for (int k0 = 0; k0 < 256; k0 += 32) {
      __syncthreads();
#pragma unroll
      for (int i = 0; i < 8; ++i) {
        int idx = tid + i * 256;
        int ar = idx >> 5, ac = idx & 31;
        int row = rb + ar;
        int bb = row >> 8, nn = row & 255;
        float xv = X[((bb * 32 + l) * 256 + nn) * 256 + k0 + ac];
        if (seg) xv *= wsm[nn * 256 + k0 + ac];
        As[ar * LDA + ac] = xv;
        int bk = idx >> 6, bc = idx & 63;      // G already B-layout (k-major, 1024 cols)
        Bs[bk * LDB + bc] = G[(k0 + bk) * 1024 + cb + bc];
      }
      __syncthreads();
      mma_chunk(As, Bs, wr, cp, r, half, acc0, acc1);
    }
  }
  for (int k0 = 0; k0 < 256; k0 += 32) {       // h @ Whh^T
    __syncthreads();
#pragma unroll
    for (int i = 0; i < 8; ++i) {
      int idx = tid + i * 256;
      int ar = idx >> 5, ac = idx & 31;
      As[ar * LDA + ac] = h[(rb + ar) * 256 + k0 + ac];
      int bc = idx >> 5, bk = idx & 31;
      Bs[bk * LDB + bc] = Whh[(cb + bc) * 256 + k0 + bk];
    }
    __syncthreads();
    mma_chunk(As, Bs, wr, cp, r, half, acc0, acc1);
  }
  float* o0 = z + (rb + 16 * wr) * 1024 + cb + 32 * cp;
  store_tile(o0, 1024, acc0, r, half);
  store_tile(o0 + 16, 1024, acc1, r, half);
}

// ---- K3 (per step l): LSTM cell pointwise; gate order i,f,g,o ----
__global__ void lstm_point(const float* __restrict__ z, const float* __restrict__ bih,
                           const float* __restrict__ bhh, float* __restrict__ h,
                           float* __restrict__ c, float* __restrict__ temp, int l) {
  int idx = blockIdx.x * 256 + threadIdx.x;    // (row 0..2047, m 0..255)
  int row = idx >> 8, m = idx & 255;
  const float* zr = z + row * 1024;
  float iv = sigmoidf(zr[m]       + bih[m]       + bhh[m]);
  float fv = sigmoidf(zr[m + 256] + bih[m + 256] + bhh[m + 256]);
  float gv = tanhf   (zr[m + 512] + bih[m + 512] + bhh[m + 512]);
  float ov = sigmoidf(zr[m + 768] + bih[m + 768] + bhh[m + 768]);
  float cv = fv * c[idx] + iv * gv;
  float hv = ov * tanhf(cv);
  c[idx] = cv;
  h[idx] = hv;
  int bb = row >> 8, nn = row & 255;
  temp[((bb * 32 + l) * 256 + nn) * 256 + m] = hv;
}

__global__ void zero_f32(float* __restrict__ p, int n) {
  int i = blockIdx.x * 256 + threadIdx.x;
  if (i < n) p[i] = 0.0f;
}

// ---- K4/K5: BatchNorm stats over 65536 samples per channel ----
__global__ void bn_stats(const float* __restrict__ temp, float* __restrict__ stats) {
  const int m = threadIdx.x;
  const int rb = blockIdx.x * 256;
  float s = 0.0f, s2 = 0.0f;
  for (int rr = 0; rr < 256; ++rr) {
    float v = temp[(rb + rr) * 256 + m];
    s += v; s2 += v * v;
  }
  atomicAdd(&stats[m], s);
  atomicAdd(&stats[256 + m], s2);
}

__global__ void bn_final(const float* __restrict__ stats, const float* __restrict__ gamma,
                         const float* __restrict__ beta, float* __restrict__ scale,
                         float* __restrict__ shift) {
  const int m = threadIdx.x;
  const float inv = 1.0f / 65536.0f;
  float mu = stats[m] * inv;
  float var = stats[256 + m] * inv - mu * mu;
  float sc = gamma[m] * rsqrtf(var + 1e-5f);
  scale[m] = sc;
  shift[m] = beta[m] - mu * sc;
}

// ---- K6: proj = BN(temp) @ Wlin^T + blin   (65536 x 256, K=256), BN fused on A load ----
__global__ void proj_gemm(const float* __restrict__ temp, const float* __restrict__ Wlin,
                          const float* __restrict__ blin, const float* __restrict__ scale,
                          const float* __restrict__ shift, float* __restrict__ proj) {
  __shared__ float As[64 * LDA];
  __shared__ float Bs[32 * LDB];
  WMMA_THREAD_DECOMP();
  const int rb = blockIdx.y * 64;
  const int cb = blockIdx.x * 64;
  v8f acc0 = {}; v8f acc1 = {};
  for (int k0 = 0; k0 < 256; k0 += 32) {
    __syncthreads();
#pragma unroll
    for (int i = 0; i < 8; ++i) {
      int idx = tid + i * 256;
      int ar = idx >> 5, ac = idx & 31;
      As[ar * LDA + ac] = temp[(rb + ar) * 256 + k0 + ac] * scale[k0 + ac] + shift[k0 + ac];
      int bc = idx >> 5, bk = idx & 31;
      Bs[bk * LDB + bc] = Wlin[(cb + bc) * 256 + k0 + bk];
    }
    __syncthreads();
    mma_chunk(As, Bs, wr, cp, r, half, acc0, acc1);
  }
  const int c0 = cb + 32 * cp;
  float* o0 = proj + (rb + 16 * wr) * 256 + c0;
#pragma unroll
  for (int i = 0; i < 8; ++i) {
    o0[(i + 8 * half) * 256 + r]      = acc0[i] + blin[c0 + r];
    o0[(i + 8 * half) * 256 + 16 + r] = acc1[i] + blin[c0 + 16 + r];
  }
}

// ---- K7: out[bl][o][d] = sum_m proj[bl][o][m] * BN(temp[bl][d][m]), BN fused on B load ----
__global__ void out_gemm(const float* __restrict__ proj, const float* __restrict__ temp,
                         const float* __restrict__ scale, const float* __restrict__ shift,
                         float* __restrict__ out) {
  __shared__ float As[64 * LDA];
  __shared__ float Bs[32 * LDB];
  WMMA_THREAD_DECOMP();
  const int bl = blockIdx.z;
  const int rb = blockIdx.y * 64;
  const int cb = blockIdx.x * 64;
  const float* P = proj + bl * 65536;
  const float* T = temp + bl * 65536;
  v8f acc0 = {}; v8f acc1 = {};
  for (int k0 = 0; k0 < 256; k0 += 32) {
    __syncthreads();
#pragma unroll
    for (int i = 0; i < 8; ++i) {
      int idx = tid + i * 256;
      int ar = idx >> 5, ac = idx & 31;
      As[ar * LDA + ac] = P[(rb + ar) * 256 + k0 + ac];
      int dd = idx >> 5, mm = idx & 31;                      // B[m][d] = BN(T[d][m])
      Bs[mm * LDB + dd] = T[(cb + dd) * 256 + k0 + mm] * scale[k0 + mm] + shift[k0 + mm];
    }
    __syncthreads();
    mma_chunk(As, Bs, wr, cp, r, half, acc0, acc1);
  }
  float* o0 = out + bl * 65536 + (rb + 16 * wr) * 256 + cb + 32 * cp;
  store_tile(o0, 256, acc0, r, half);
  store_tile(o0 + 16, 256, acc1, r, half);
}

extern "C" void kernel_launch(void* const* d_in, const int* in_sizes, int n_in,
                              void* d_out, int out_size, void* d_ws, size_t ws_size,
                              hipStream_t stream) {
  (void)in_sizes; (void)n_in; (void)out_size; (void)ws_size;
  const float* X     = (const float*)d_in[0];
  const float* dis   = (const float*)d_in[1];
  const float* E1    = (const float*)d_in[2];
  const float* E2    = (const float*)d_in[3];
  const float* Wih   = (const float*)d_in[4];
  const float* Whh   = (const float*)d_in[5];
  const float* bih   = (const float*)d_in[6];
  const float* bhh   = (const float*)d_in[7];
  const float* gamma = (const float*)d_in[8];
  const float* beta  = (const float*)d_in[9];
  const float* Wlin  = (const float*)d_in[10];
  const float* blin  = (const float*)d_in[11];
  float* out = (float*)d_out;

  // workspace layout (floats); total ~37.4M floats ~ 150 MB
  float* ws = (float*)d_ws;
  size_t o = 0;
  float* w_sm  = ws + o; o += 256 * 256;      // softmax weights
  float* G1    = ws + o; o += 256 * 1024;     // E1 @ Wih[:, :256]^T
  float* G2    = ws + o; o += 256 * 1024;     // E2 @ Wih[:, 256:]^T
  float* z     = ws + o; o += 2048 * 1024;    // per-step gate pre-activations
  float* h     = ws + o; o += 2048 * 256;     // LSTM hidden
  float* c     = ws + o; o += 2048 * 256;     // LSTM cell  (contiguous after h)
  float* stats = ws + o; o += 512;            // per-channel sum / sumsq (contiguous after c)
  float* scale = ws + o; o += 256;            // BN fused scale
  float* shift = ws + o; o += 256;            // BN fused shift
  float* temp  = ws + o; o += 65536 * 256;    // all hidden states (B,L,N,M)
  float* proj  = ws + o; o += 65536 * 256;    // tb @ Wlin^T + blin

  softmax_rows<<<256, 256, 0, stream>>>(dis, w_sm);
  embed_weight_gemm<<<dim3(16, 4, 2), 256, 0, stream>>>(E1, E2, Wih, G1, G2);

  const int zn = 2048 * 256 * 2 + 512;        // h, c, stats are contiguous
  zero_f32<<<(zn + 255) / 256, 256, 0, stream>>>(h, zn);

  for (int l = 0; l < 32; ++l) {
    lstm_gemm<<<dim3(16, 32), 256, 0, stream>>>(X, w_sm, G1, G2, Whh, h, z, l);
    lstm_point<<<2048, 256, 0, stream>>>(z, bih, bhh, h, c, temp, l);
  }

  bn_stats<<<256, 256, 0, stream>>>(temp, stats);
  bn_final<<<1, 256, 0, stream>>>(stats, gamma, beta, scale, shift);
  proj_gemm<<<dim3(4, 1024), 256, 0, stream>>>(temp, Wlin, blin, scale, shift, proj);
  out_gemm<<<dim3(4, 4, 256), 256, 0, stream>>>(proj, temp, scale, shift, out);
}